// Top2Gate_12498354831551
// MI455X (gfx1250) — compile-verified
//
#include <hip/hip_runtime.h>
#include <hip/hip_bf16.h>

#define S_TOK 8192
#define E_EXP 64
#define KDIM  2048
#define CAP   256
#define KC    64          // K-chunk staged in LDS
#define KCP   (KC + 4)    // padded row stride (floats) to avoid bank conflicts
#define NELT  ((long long)S_TOK * E_EXP * CAP)   // 134217728 per output tensor

typedef float v2f __attribute__((ext_vector_type(2)));
typedef float v8f __attribute__((ext_vector_type(8)));

// ---------------------------------------------------------------------------
// Zero-fill the full output (1 + 2*NELT floats) with wide stores.
// ---------------------------------------------------------------------------
__global__ void k_zero(float4* __restrict__ p, long long nq, float* __restrict__ tail) {
    long long i = (long long)blockIdx.x * blockDim.x + threadIdx.x;
    const long long stride = (long long)gridDim.x * blockDim.x;
    const float4 z = make_float4(0.f, 0.f, 0.f, 0.f);
    for (; i < nq; i += stride) p[i] = z;
    if (blockIdx.x == 0 && threadIdx.x == 0) *tail = 0.f;
}

__global__ void k_init(float* __restrict__ me_sum) {
    me_sum[threadIdx.x] = 0.f;
}

// ---------------------------------------------------------------------------
// Fused: logits = x @ w^T via V_WMMA_F32_16X16X4_F32, then softmax + top-2.
// Block = 128 threads (4 waves), each block owns 64 tokens x 64 experts.
// Wave w computes rows [16w, 16w+16); 4 N-tiles of 16 experts each.
// ---------------------------------------------------------------------------
__global__ __launch_bounds__(128) void k_gate_gemm(
    const float* __restrict__ x, const float* __restrict__ w,
    float* __restrict__ g1raw, float* __restrict__ g2raw,
    int* __restrict__ idx1, int* __restrict__ idx2,
    float* __restrict__ me_sum)
{
    __shared__ float xs[64 * KCP];
    __shared__ float wsld[64 * KCP];
    __shared__ float lg[64 * 65];     // logits tile, padded stride 65
    __shared__ float maxv[64];
    __shared__ float sumv[64];

    const int tid  = threadIdx.x;
    const int wave = tid >> 5;
    const int lane = tid & 31;
    const int tok0 = blockIdx.x * 64;

    v8f acc0 = {}, acc1 = {}, acc2 = {}, acc3 = {};

    const int mrow  = (wave << 4) + (lane & 15);   // A row this lane feeds
    const int khalf = (lane >> 4) << 1;            // lanes 0-15: K+0/1, 16-31: K+2/3
    const int ncol  = lane & 15;                   // B column within an N-tile

    for (int k0 = 0; k0 < KDIM; k0 += KC) {
        __syncthreads();
        // Stage x[64 x KC] and w[64 x KC] into LDS with float4 loads.
        for (int i = tid; i < 64 * (KC / 4); i += 128) {
            const int r  = i >> 4;        // KC/4 = 16 float4 per row
            const int c4 = (i & 15) << 2;
            *reinterpret_cast<float4*>(&xs[r * KCP + c4]) =
                *reinterpret_cast<const float4*>(x + (size_t)(tok0 + r) * KDIM + k0 + c4);
            *reinterpret_cast<float4*>(&wsld[r * KCP + c4]) =
                *reinterpret_cast<const float4*>(w + (size_t)r * KDIM + k0 + c4);
        }
        if (k0 + KC < KDIM) {   // global_prefetch_b8 of the next chunk
            __builtin_prefetch(x + (size_t)(tok0 + (tid >> 1)) * KDIM + k0 + KC + (tid & 1) * 32, 0, 1);
            __builtin_prefetch(w + (size_t)(tid >> 1) * KDIM + k0 + KC + (tid & 1) * 32, 0, 1);
        }
        __syncthreads();

        for (int kk = 0; kk < KC; kk += 4) {
            v2f a;
            a.x = xs[mrow * KCP + kk + khalf];
            a.y = xs[mrow * KCP + kk + khalf + 1];
            v2f b0, b1, b2, b3;
            b0.x = wsld[(ncol +  0) * KCP + kk + khalf];
            b0.y = wsld[(ncol +  0) * KCP + kk + khalf + 1];
            b1.x = wsld[(ncol + 16) * KCP + kk + khalf];
            b1.y = wsld[(ncol + 16) * KCP + kk + khalf + 1];
            b2.x = wsld[(ncol + 32) * KCP + kk + khalf];
            b2.y = wsld[(ncol + 32) * KCP + kk + khalf + 1];
            b3.x = wsld[(ncol + 48) * KCP + kk + khalf];
            b3.y = wsld[(ncol + 48) * KCP + kk + khalf + 1];
            acc0 = __builtin_amdgcn_wmma_f32_16x16x4_f32(false, a, false, b0, (short)0, acc0, false, false);
            acc1 = __builtin_amdgcn_wmma_f32_16x16x4_f32(false, a, false, b1, (short)0, acc1, false, false);
            acc2 = __builtin_amdgcn_wmma_f32_16x16x4_f32(false, a, false, b2, (short)0, acc2, false, false);
            acc3 = __builtin_amdgcn_wmma_f32_16x16x4_f32(false, a, false, b3, (short)0, acc3, false, false);
        }
    }

    // Spill C tiles to LDS. C layout: VGPR r -> M=r (lanes 0-15), M=r+8 (16-31).
    union U { v8f v; float f[8]; };
    U u0, u1, u2, u3; u0.v = acc0; u1.v = acc1; u2.v = acc2; u3.v = acc3;
    const int mbase = (wave << 4) + ((lane >> 4) << 3);
    const int nlo   = lane & 15;
#pragma unroll
    for (int r = 0; r < 8; ++r) {
        lg[(mbase + r) * 65 + nlo +  0] = u0.f[r];
        lg[(mbase + r) * 65 + nlo + 16] = u1.f[r];
        lg[(mbase + r) * 65 + nlo + 32] = u2.f[r];
        lg[(mbase + r) * 65 + nlo + 48] = u3.f[r];
    }
    __syncthreads();

    // Per-token softmax + top-2 (threads 0..63, one token each).
    if (tid < 64) {
        const int t = tid;
        float mx = -3.402823466e38f; int i1 = 0;
        for (int e = 0; e < 64; ++e) { float v = lg[t * 65 + e]; if (v > mx) { mx = v; i1 = e; } }
        float mx2 = -3.402823466e38f; int i2 = 0;
        for (int e = 0; e < 64; ++e) {
            if (e == i1) continue;
            float v = lg[t * 65 + e]; if (v > mx2) { mx2 = v; i2 = e; }
        }
        float s = 0.f;
        for (int e = 0; e < 64; ++e) s += __expf(lg[t * 65 + e] - mx);
        maxv[t] = mx; sumv[t] = s;
        const int token = tok0 + t;
        g1raw[token] = 1.0f / s;                 // exp(0)/s
        g2raw[token] = __expf(mx2 - mx) / s;
        idx1[token] = i1; idx2[token] = i2;
    }
    __syncthreads();

    // Per-expert gate-mean accumulation (transposed read: one expert per thread).
    if (tid < 64) {
        const int e = tid;
        float acc = 0.f;
        for (int t = 0; t < 64; ++t) acc += __expf(lg[t * 65 + e] - maxv[t]) / sumv[t];
        atomicAdd(&me_sum[e], acc);
    }
}

// ---------------------------------------------------------------------------
// Per-chunk expert histograms (32 chunks of 256 tokens).
// ---------------------------------------------------------------------------
__global__ __launch_bounds__(256) void k_hist(const int* __restrict__ idx1, const int* __restrict__ idx2,
                                              int* __restrict__ hist1, int* __restrict__ hist2) {
    __shared__ int h1[64], h2[64];
    const int tid = threadIdx.x;
    if (tid < 64) { h1[tid] = 0; h2[tid] = 0; }
    __syncthreads();
    const int token = blockIdx.x * 256 + tid;
    atomicAdd(&h1[idx1[token]], 1);
    atomicAdd(&h2[idx2[token]], 1);
    __syncthreads();
    if (tid < 64) { hist1[blockIdx.x * 64 + tid] = h1[tid]; hist2[blockIdx.x * 64 + tid] = h2[tid]; }
}

// ---------------------------------------------------------------------------
// Exclusive prefix over chunks per expert; totals; l_aux.
// ---------------------------------------------------------------------------
__global__ __launch_bounds__(64) void k_scan(const int* __restrict__ hist1, const int* __restrict__ hist2,
                                             int* __restrict__ off1, int* __restrict__ off2,
                                             int* __restrict__ cnt1,
                                             const float* __restrict__ me_sum, float* __restrict__ d_out) {
    __shared__ float red[64];
    const int e = threadIdx.x;
    int run1 = 0, run2 = 0;
    for (int b = 0; b < 32; ++b) {
        off1[b * 64 + e] = run1; run1 += hist1[b * 64 + e];
        off2[b * 64 + e] = run2; run2 += hist2[b * 64 + e];
    }
    cnt1[e] = run1;   // sum(mask1, axis=0): pre-drop count, feeds ce and loc2 base
    red[e] = (me_sum[e] / (float)S_TOK) * ((float)run1 / (float)S_TOK);
    __syncthreads();
    if (e == 0) {
        float s = 0.f;
        for (int i = 0; i < 64; ++i) s += red[i];
        d_out[0] = s * (float)E_EXP;   // mean(me*ce)*E*E == sum*E
    }
}

// ---------------------------------------------------------------------------
// Within-chunk rank + final locations.
// ---------------------------------------------------------------------------
__global__ __launch_bounds__(256) void k_rank(const int* __restrict__ idx1, const int* __restrict__ idx2,
                                              const int* __restrict__ off1, const int* __restrict__ off2,
                                              const int* __restrict__ cnt1,
                                              int* __restrict__ loc1, int* __restrict__ loc2) {
    __shared__ int s1[256], s2[256];
    const int tid = threadIdx.x;
    const int b = blockIdx.x;
    const int token = b * 256 + tid;
    const int e1 = idx1[token], e2 = idx2[token];
    s1[tid] = e1; s2[tid] = e2;
    __syncthreads();
    int r1 = 0, r2 = 0;
    for (int t = 0; t < tid; ++t) { r1 += (s1[t] == e1); r2 += (s2[t] == e2); }
    loc1[token] = off1[b * 64 + e1] + r1;
    loc2[token] = off2[b * 64 + e2] + r2 + cnt1[e2];
}

// ---------------------------------------------------------------------------
// Capacity drop + renormalize + scatter into combine_weights / dispatch_mask.
// ---------------------------------------------------------------------------
__global__ __launch_bounds__(256) void k_scatter(const int* __restrict__ idx1, const int* __restrict__ idx2,
                                                 const int* __restrict__ loc1, const int* __restrict__ loc2,
                                                 const float* __restrict__ g1raw, const float* __restrict__ g2raw,
                                                 float* __restrict__ cw, float* __restrict__ dm) {
    const int token = blockIdx.x * 256 + threadIdx.x;
    const int e1 = idx1[token], e2 = idx2[token];
    const int l1 = loc1[token], l2 = loc2[token];
    const bool keep1 = l1 < CAP, keep2 = l2 < CAP;
    float g1 = keep1 ? g1raw[token] : 0.f;
    float g2 = keep2 ? g2raw[token] : 0.f;
    const float denom = fmaxf(g1 + g2, 1.1920929e-07f);
    g1 /= denom; g2 /= denom;
    const size_t base = (size_t)token * E_EXP * CAP;
    if (keep1) {
        const size_t o = base + (size_t)e1 * CAP + l1;
        cw[o] = g1; dm[o] = (g1 != 0.f) ? 1.f : 0.f;
    }
    if (keep2) {
        const size_t o = base + (size_t)e2 * CAP + l2;
        cw[o] = g2; dm[o] = (g2 != 0.f) ? 1.f : 0.f;
    }
}

// ---------------------------------------------------------------------------
extern "C" void kernel_launch(void* const* d_in, const int* in_sizes, int n_in,
                              void* d_out, int out_size, void* d_ws, size_t ws_size,
                              hipStream_t stream) {
    const float* x = (const float*)d_in[0];       // [8192, 2048]
    const float* w = (const float*)d_in[1];       // [64, 2048]
    float* out = (float*)d_out;                   // 1 + 2*NELT floats

    // Workspace layout
    float* me_sum = (float*)d_ws;                 // 64
    float* g1raw  = me_sum + 64;                  // 8192
    float* g2raw  = g1raw + S_TOK;                // 8192
    int*   idx1   = (int*)(g2raw + S_TOK);        // 8192
    int*   idx2   = idx1 + S_TOK;
    int*   loc1   = idx2 + S_TOK;
    int*   loc2   = loc1 + S_TOK;
    int*   hist1  = loc2 + S_TOK;                 // 32*64
    int*   hist2  = hist1 + 32 * 64;
    int*   off1   = hist2 + 32 * 64;
    int*   off2   = off1 + 32 * 64;
    int*   cnt1   = off2 + 32 * 64;               // 64

    const long long total = 1 + 2 * NELT;         // 268435457 floats
    const long long nq = total / 4;               // float4 count (covers 0..total-2)

    k_zero<<<4096, 256, 0, stream>>>((float4*)out, nq, out + (total - 1));
    k_init<<<1, 64, 0, stream>>>(me_sum);
    k_gate_gemm<<<S_TOK / 64, 128, 0, stream>>>(x, w, g1raw, g2raw, idx1, idx2, me_sum);
    k_hist<<<32, 256, 0, stream>>>(idx1, idx2, hist1, hist2);
    k_scan<<<1, 64, 0, stream>>>(hist1, hist2, off1, off2, cnt1, me_sum, out);
    k_rank<<<32, 256, 0, stream>>>(idx1, idx2, off1, off2, cnt1, loc1, loc2);
    k_scatter<<<32, 256, 0, stream>>>(idx1, idx2, loc1, loc2, g1raw, g2raw,
                                      out + 1, out + 1 + NELT);
}